// GLIF_13881334300839
// MI455X (gfx1250) — compile-verified
//
#include <hip/hip_runtime.h>
#include <math.h>

typedef float v2f __attribute__((ext_vector_type(2)));
typedef float v8f __attribute__((ext_vector_type(8)));

#define NN 4096
#define T_STEPS 2048
#define ROWS_PER_BLOCK 16
#define WAVES_PER_BLOCK 8
#define THREADS (WAVES_PER_BLOCK * 32)
#define K_PER_WAVE (NN / WAVES_PER_BLOCK)   // 512
#define QUADS_PER_WAVE (K_PER_WAVE / 4)     // 128
#define QUADS_TOTAL (NN / 4)                // 1024 quads per row-tile

// One-time state initialization from the reference's initial-condition inputs.
__global__ void glif_init(const float* __restrict__ v0, const float* __restrict__ ts0,
                          const float* __restrict__ tv0, const float* __restrict__ ia0,
                          float* __restrict__ v, float* __restrict__ ts,
                          float* __restrict__ tv, float* __restrict__ ia) {
  int i = blockIdx.x * blockDim.x + threadIdx.x;
  if (i < NN) { v[i] = v0[i]; ts[i] = ts0[i]; tv[i] = tv0[i]; ia[i] = ia0[i]; }
}

// One-time: repack w into WMMA-A-fragment-contiguous order so each step's
// A-load is a single fully coalesced 256B wave transaction.
// packed[(rt*QUADS_TOTAL + q)*32 + lane] = float2{ w[rt*16 + (lane&15)][4q + 2*(lane>>4) + 0],
//                                                 w[...][... + 1] }
__global__ void glif_pack_w(const float* __restrict__ w, float2* __restrict__ packed) {
  const unsigned flat = blockIdx.x * blockDim.x + threadIdx.x;  // < 256*1024*32
  const int lane = flat & 31;
  const int q    = (flat >> 5) & (QUADS_TOTAL - 1);
  const int rt   = flat >> 15;
  const int row  = rt * ROWS_PER_BLOCK + (lane & 15);
  const int col  = q * 4 + ((lane >> 4) << 1);
  packed[flat] = *(const float2*)(w + (size_t)row * NN + col);
}

// One timestep: I = x_t + w @ I_add  (WMMA f32 16x16x4), then GLIF elementwise update.
template <bool PACKED>
__launch_bounds__(THREADS)
__global__ void glif_step(int t,
    const float* __restrict__ x_in, const float* __restrict__ wmat,
    const float* __restrict__ E_L,  const float* __restrict__ C_m,
    const float* __restrict__ G,    const float* __restrict__ R_I,
    const float* __restrict__ f_v,  const float* __restrict__ f_I,
    const float* __restrict__ dth_s,const float* __restrict__ b_s,
    const float* __restrict__ a_v,  const float* __restrict__ b_v,
    const float* __restrict__ th_inf,const float* __restrict__ dV,
    const float* __restrict__ I_A,
    float* __restrict__ v_st, float* __restrict__ ts_st, float* __restrict__ tv_st,
    const float* __restrict__ ia_rd, float* __restrict__ ia_wr,
    float* __restrict__ out)
{
  __shared__ float lds_I[NN];                              // 16 KB: full I_add vector
  __shared__ float red[WAVES_PER_BLOCK][ROWS_PER_BLOCK];   // cross-wave partials

  const int tid  = threadIdx.x;
  const int wave = tid >> 5;
  const int lane = tid & 31;
  const int m0   = blockIdx.x * ROWS_PER_BLOCK;

  // Stage I_add into LDS, fully coalesced (float4).
  {
    const float4* src4 = (const float4*)ia_rd;
    float4* dst4 = (float4*)lds_I;
#pragma unroll
    for (int i = 0; i < NN / 4 / THREADS; ++i)
      dst4[tid + i * THREADS] = src4[tid + i * THREADS];
  }
  __syncthreads();

  // --- WMMA matvec over this wave's K slice ------------------------------
  // A fragment (16x4 f32): lane&15 = row M, lane>>4 selects K-half {0,1}/{2,3}.
  // B fragment (4x16 f32): same K mapping, value replicated across all N cols.
  const int khalf  = (lane >> 4) << 1;          // 0 or 2
  const int kstart = wave * K_PER_WAVE;

  v8f c = {0.f, 0.f, 0.f, 0.f, 0.f, 0.f, 0.f, 0.f};

  if (PACKED) {
    // Fragment-contiguous: each iteration is one coalesced 256B wave load.
    const v2f* __restrict__ base = (const v2f*)wmat +
        ((size_t)blockIdx.x * QUADS_TOTAL + (size_t)wave * QUADS_PER_WAVE) * 32 + lane;
    for (int q = 0; q < QUADS_PER_WAVE; ++q) {
      v2f a = base[(size_t)q * 32];                    // global_load_b64, coalesced
      v2f b = *(const v2f*)(lds_I + kstart + q * 4 + khalf);
      c = __builtin_amdgcn_wmma_f32_16x16x4_f32(
              false, a, false, b, (short)0, c, false, false);
    }
  } else {
    const float* wrow = wmat + (size_t)(m0 + (lane & 15)) * NN;
    for (int q = 0; q < QUADS_PER_WAVE; ++q) {
      const int kb = kstart + q * 4 + khalf;
      v2f a = *(const v2f*)(wrow + kb);
      v2f b = *(const v2f*)(lds_I + kb);
      c = __builtin_amdgcn_wmma_f32_16x16x4_f32(
              false, a, false, b, (short)0, c, false, false);
    }
  }

  // All 16 D columns are identical. Lane 0 holds rows 0..7, lane 16 rows 8..15.
  if (lane == 0) {
#pragma unroll
    for (int m = 0; m < 8; ++m) red[wave][m] = c[m];
  } else if (lane == 16) {
#pragma unroll
    for (int m = 0; m < 8; ++m) red[wave][8 + m] = c[m];
  }
  __syncthreads();

  // --- Deterministic cross-wave reduction + GLIF elementwise update ------
  if (tid < ROWS_PER_BLOCK) {
    const int n = m0 + tid;
    float acc = 0.f;
#pragma unroll
    for (int wv = 0; wv < WAVES_PER_BLOCK; ++wv) acc += red[wv][tid];

    const float I    = x_in[(size_t)t * NN + n] + acc;
    const float v    = v_st[n];
    const float ths  = ts_st[n];
    const float thv  = tv_st[n];
    const float ia   = ia_rd[n];
    const float el   = E_L[n];

    const float dv      = (I * R_I[n] - G[n] * (v - el)) / C_m[n];
    const float v_next  = v + dv;
    const float thr     = ths + thv;
    const float ssoft   = 1.f / (1.f + expf(-(v_next - thr)));
    const float sp      = (v_next >= thr) ? 1.f : 0.f;
    const float nsp     = 1.f - sp;
    const float v_reset = el + f_v[n] * (v - el) - dV[n];
    const float v_new   = sp * v_reset + nsp * v_next;
    const float ths_new = (1.f - b_s[n]) * ths + sp * dth_s[n];
    const float dthv    = a_v[n] * (v_new - el) - b_v[n] * (thv - th_inf[n]);
    const float thv_new = thv + nsp * dthv;
    const float ia_new  = (1.f - f_I[n]) * ia + ssoft * I_A[n];

    v_st[n]  = v_new;
    ts_st[n] = ths_new;
    tv_st[n] = thv_new;
    ia_wr[n] = ia_new;

    out[(size_t)t * NN + n]             = v_new;   // v_seq  at [0, t, n]
    out[(size_t)(T_STEPS + t) * NN + n] = ssoft;   // s_seq  at [1, t, n]
  }
}

extern "C" void kernel_launch(void* const* d_in, const int* in_sizes, int n_in,
                              void* d_out, int out_size, void* d_ws, size_t ws_size,
                              hipStream_t stream) {
  const float* x_in   = (const float*)d_in[0];
  const float* w      = (const float*)d_in[1];
  const float* E_L    = (const float*)d_in[2];
  const float* C_m    = (const float*)d_in[3];
  const float* G      = (const float*)d_in[4];
  const float* R_I    = (const float*)d_in[5];
  const float* f_v    = (const float*)d_in[6];
  const float* f_I    = (const float*)d_in[7];
  const float* dth_s  = (const float*)d_in[8];
  const float* b_s    = (const float*)d_in[9];
  const float* a_v    = (const float*)d_in[10];
  const float* b_v    = (const float*)d_in[11];
  const float* th_inf = (const float*)d_in[12];
  const float* dV     = (const float*)d_in[13];
  const float* I_A    = (const float*)d_in[14];
  const float* v0     = (const float*)d_in[15];
  const float* ts0    = (const float*)d_in[16];
  const float* tv0    = (const float*)d_in[17];
  const float* ia0    = (const float*)d_in[18];

  float* ws  = (float*)d_ws;
  float* iaA = ws + 0 * NN;   // I_add ping
  float* iaB = ws + 1 * NN;   // I_add pong
  float* v   = ws + 2 * NN;
  float* ts  = ws + 3 * NN;
  float* tv  = ws + 4 * NN;
  float* wpk = ws + 5 * NN;   // packed w (64 MB), if workspace permits

  const size_t need_packed = ((size_t)5 * NN + (size_t)NN * NN) * sizeof(float);
  const bool use_packed = (ws_size >= need_packed);

  glif_init<<<(NN + 255) / 256, 256, 0, stream>>>(v0, ts0, tv0, ia0, v, ts, tv, iaA);

  if (use_packed) {
    // 256 row-tiles * 1024 quads * 32 lanes = 8M threads, one float2 each.
    const unsigned total = 256u * QUADS_TOTAL * 32u;
    glif_pack_w<<<total / 256, 256, 0, stream>>>(w, (float2*)wpk);
  }

  for (int t = 0; t < T_STEPS; ++t) {
    const float* rd = (t & 1) ? iaB : iaA;
    float*       wr = (t & 1) ? iaA : iaB;
    if (use_packed) {
      glif_step<true><<<NN / ROWS_PER_BLOCK, THREADS, 0, stream>>>(
          t, x_in, wpk, E_L, C_m, G, R_I, f_v, f_I, dth_s, b_s, a_v, b_v,
          th_inf, dV, I_A, v, ts, tv, rd, wr, (float*)d_out);
    } else {
      glif_step<false><<<NN / ROWS_PER_BLOCK, THREADS, 0, stream>>>(
          t, x_in, w, E_L, C_m, G, R_I, f_v, f_I, dth_s, b_s, a_v, b_v,
          th_inf, dV, I_A, v, ts, tv, rd, wr, (float*)d_out);
    }
  }
}